// DistanceTransform_91285234909315
// MI455X (gfx1250) — compile-verified
//
#include <hip/hip_runtime.h>
#include <hip/hip_bf16.h>
#include <math.h>

// Problem constants (B=2, C=3, H=256, W=256), INF = (H+W)^2
#define IMGS 6
#define HN 256
#define WN 256
#define INF_F 262144.0f

#ifndef __has_builtin
#define __has_builtin(x) 0
#endif

#if __has_builtin(__builtin_amdgcn_global_load_async_to_lds_b32)
#define HAVE_ASYNC_LDS 1
#else
#define HAVE_ASYNC_LDS 0
#endif

typedef __attribute__((address_space(1))) void gas_void;
typedef __attribute__((address_space(3))) void las_void;
typedef __attribute__((address_space(1))) int gas_int;
typedef __attribute__((address_space(3))) int las_int;

typedef __attribute__((ext_vector_type(2))) float v2f;
typedef __attribute__((ext_vector_type(8))) float v8f;

// Async DMA one dword per lane: global -> LDS. Tracked with ASYNCcnt on CDNA5.
__device__ __forceinline__ void copy_g2l_b32(const float* gsrc, float* ldst) {
#if HAVE_ASYNC_LDS
  gas_void* gv = (gas_void*)gsrc;  // addrspacecast generic -> global
  las_void* lv = (las_void*)ldst;  // addrspacecast generic -> LDS
  __builtin_amdgcn_global_load_async_to_lds_b32(
      (gas_int*)gv, (las_int*)lv, /*offset=*/0, /*cpol=*/0);
#else
  *ldst = *gsrc;
#endif
}

__device__ __forceinline__ void wait_async0() {
#if HAVE_ASYNC_LDS
#if __has_builtin(__builtin_amdgcn_s_wait_asynccnt)
  __builtin_amdgcn_s_wait_asynccnt(0);
#else
  asm volatile("s_wait_asynccnt 0" ::: "memory");
#endif
#endif
}

// One 16(p) x 16(q) tile of the min-plus DT via V_WMMA_F32_16X16X4_F32:
//   D[m][n] = t[q0+m] + (-2*(q0+m))*(p0+n) + (p0+n)^2  ==  (p-q)^2 + f[q]
// A (16x4, M=q): lanes 0-15 hold K0,K1 = {t[q], -2q}; lanes 16-31 hold K2,K3 = {1, 0}.
// B (4x16,  N=p): lanes 0-15 hold K0,K1 = {1, p};     lanes 16-31 hold K2,K3 = {p^2, 0}.
// All terms are integers < 2^24 -> exact fp32, matching the reference bit-for-bit.
__device__ __forceinline__ v8f dt_tile_wmma(float tq, int q, int p, int half) {
  v2f av, bv;
  float qf = (float)q;
  float pf = (float)p;
  av.x = half ? 1.0f : tq;
  av.y = half ? 0.0f : (-2.0f * qf);
  bv.x = half ? (pf * pf) : 1.0f;
  bv.y = half ? 0.0f : pf;
  v8f c = {};
  return __builtin_amdgcn_wmma_f32_16x16x4_f32(
      /*neg_a=*/false, av, /*neg_b=*/false, bv,
      /*c_mod=*/(short)0, c, /*reuse_a=*/false, /*reuse_b=*/false);
}

// ---------------------------------------------------------------------------
// Pass 1: 1D min-plus DT along W. 192 blocks x (32,8); each wave32 owns one row.
// Async-stage the 1KB row to LDS, build t[q] = q^2 + {0,INF}, then 16 p-tiles
// x 16 q-tiles of WMMA + per-lane v_min; one shfl_xor(16) merges wave halves.
// ---------------------------------------------------------------------------
__global__ __launch_bounds__(256) void dt_pass1_w(const float* __restrict__ x,
                                                  float* __restrict__ g) {
  __shared__ float t[8][WN];
  const int tx = threadIdx.x;   // 0..31 (lane, wave32)
  const int ty = threadIdx.y;   // 0..7  (wave id)
  const int row = blockIdx.x * 8 + ty;
  const float* src = x + row * WN;

#pragma unroll
  for (int k = 0; k < 8; ++k)
    copy_g2l_b32(src + tx + 32 * k, &t[ty][tx + 32 * k]);
  wait_async0();

  // t[q] = q^2 + (x==0 ? 0 : INF); NaN compares !=0 -> INF (scipy semantics)
#pragma unroll
  for (int k = 0; k < 8; ++k) {
    int q = tx + 32 * k;
    float v = t[ty][q];
    t[ty][q] = (float)(q * q) + ((v == 0.0f) ? 0.0f : INF_F);
  }

  const int l = tx & 15;
  const int half = tx >> 4;

  for (int pi = 0; pi < 16; ++pi) {
    const int p = pi * 16 + l;
    v8f acc;
#pragma unroll
    for (int j = 0; j < 8; ++j) acc[j] = 3.0e38f;

#pragma unroll 4
    for (int qi = 0; qi < 16; ++qi) {
      const int q = qi * 16 + l;
      const float tq = t[ty][q];  // lanes 0-15 feed A; upper half reads are unused
      v8f d = dt_tile_wmma(tq, q, p, half);
#pragma unroll
      for (int j = 0; j < 8; ++j) acc[j] = fminf(acc[j], d[j]);
    }

    float r = acc[0];
#pragma unroll
    for (int j = 1; j < 8; ++j) r = fminf(r, acc[j]);
    r = fminf(r, __shfl_xor(r, 16, 32));  // merge q-rows 0-7 with 8-15
    if (half == 0) g[row * WN + p] = r;
  }
}

// ---------------------------------------------------------------------------
// Pass 2: 1D min-plus DT along H + sqrt + NaN fixup. 48 blocks x (32,8).
// Stage the 256x32 column tile TRANSPOSED into padded LDS (sT[c][h], pad 260
// -> 4-bank skew, <=2-way conflicts), add h^2 to get t, then each wave runs
// 4 columns x 16 p-tiles x 16 q-tiles of WMMA + min. Results land in a second
// LDS tile; a coalesced epilogue applies sqrt + NaN and writes d_out in place.
// ---------------------------------------------------------------------------
__global__ __launch_bounds__(256) void dt_pass2_h(const float* __restrict__ x,
                                                  float* __restrict__ g) {
  __shared__ float sT[32][260];  // sT[c][h] = g[h][w0+c] + h^2
  __shared__ float rT[32][260];  // rT[c][h] = d2 result
  const int tx = threadIdx.x;
  const int ty = threadIdx.y;
  const int img = blockIdx.x >> 3;        // 0..5
  const int w0 = (blockIdx.x & 7) << 5;   // 0,32,...,224
  const float* gim = g + img * HN * WN;

#pragma unroll
  for (int k = 0; k < 32; ++k) {
    int h = ty + 8 * k;
    copy_g2l_b32(gim + h * WN + w0 + tx, &sT[tx][h]);  // transposed stage
  }
  wait_async0();
  __syncthreads();

#pragma unroll
  for (int k = 0; k < 32; ++k) {
    int h = ty + 8 * k;
    sT[tx][h] += (float)(h * h);  // t[q] = g + q^2 (exact: ints < 2^24)
  }
  __syncthreads();

  const int l = tx & 15;
  const int half = tx >> 4;
  const int c0 = ty * 4;  // each wave owns 4 columns

  for (int cc = 0; cc < 4; ++cc) {
    const int c = c0 + cc;
    for (int pi = 0; pi < 16; ++pi) {
      const int p = pi * 16 + l;  // output h index within line
      v8f acc;
#pragma unroll
      for (int j = 0; j < 8; ++j) acc[j] = 3.0e38f;

#pragma unroll 4
      for (int qi = 0; qi < 16; ++qi) {
        const int q = qi * 16 + l;
        const float tq = sT[c][q];  // contiguous -> conflict-free
        v8f d = dt_tile_wmma(tq, q, p, half);
#pragma unroll
        for (int j = 0; j < 8; ++j) acc[j] = fminf(acc[j], d[j]);
      }

      float r = acc[0];
#pragma unroll
      for (int j = 1; j < 8; ++j) r = fminf(r, acc[j]);
      r = fminf(r, __shfl_xor(r, 16, 32));
      if (half == 0) rT[c][p] = r;
    }
  }
  __syncthreads();

  const float* xim = x + img * HN * WN;
  float* gout = g + img * HN * WN;
#pragma unroll
  for (int k = 0; k < 32; ++k) {
    int h = ty + 8 * k;
    int off = h * WN + w0 + tx;
    float xv = xim[off];
    float r = sqrtf(rT[tx][h]);
    gout[off] = __builtin_isnan(xv) ? xv : r;
  }
}

extern "C" void kernel_launch(void* const* d_in, const int* in_sizes, int n_in,
                              void* d_out, int out_size, void* d_ws, size_t ws_size,
                              hipStream_t stream) {
  (void)in_sizes; (void)n_in; (void)out_size; (void)d_ws; (void)ws_size;
  const float* x = (const float*)d_in[0];
  float* out = (float*)d_out;

  dim3 blk(32, 8, 1);
  // Pass 1: B*C*H = 1536 rows, 8 rows (waves) per block.
  dt_pass1_w<<<dim3(192, 1, 1), blk, 0, stream>>>(x, out);
  // Pass 2: 6 images * 8 column tiles of 32.
  dt_pass2_h<<<dim3(48, 1, 1), blk, 0, stream>>>(x, out);
}